// EnergyEfficiencyNetTest1_6725918786374
// MI455X (gfx1250) — compile-verified
//
#include <hip/hip_runtime.h>
#include <stdint.h>

// Geometry from the reference: x is (T=1024, B=32768, 1), weights are 1x1.
#define T_STEPS 1024
#define BATCH   32768
#define BLOCK   128
#define TT      32                         // timesteps per LDS chunk
#define NCHUNK  (T_STEPS / TT)             // 32
#define ROWF    BLOCK                      // floats per chunk row (one per thread)
#define NBUF    4                          // quad-buffer: 3 chunks in flight
#define ROUNDS  ((TT * ROWF) / (4 * BLOCK))        // 8 b128 issues / thread / chunk
#define G_ROUND_STRIDE (4u * BATCH * 4u)   // each round advances 4 rows: 524288 B
#define L_ROUND_STRIDE ((unsigned)BLOCK * 16u)     // 2048 B of LDS per round

typedef __attribute__((address_space(3))) float lds_float;

// Raw 32-bit LDS byte address of a pointer into __shared__.
__device__ __forceinline__ unsigned lds_addr_of(const float* p) {
  return (unsigned)(unsigned long long)(const lds_float*)p;
}

// GVS-form async copy: uniform chunk base in SGPRs, per-thread 32-bit offset VGPR.
__device__ __forceinline__ void async_chunk(const float* xg_chunk,   // uniform
                                            unsigned lds_addr,      // per-thread LDS byte addr
                                            unsigned goff) {        // per-thread byte offset
  unsigned long long sbase = (unsigned long long)(uintptr_t)xg_chunk;
#pragma unroll
  for (int r = 0; r < ROUNDS; ++r) {
    asm volatile("global_load_async_to_lds_b128 %0, %1, %2"
                 :: "v"(lds_addr), "v"(goff), "s"(sbase)
                 : "memory");
    lds_addr += L_ROUND_STRIDE;
    goff     += G_ROUND_STRIDE;
  }
}

__device__ __forceinline__ void wait_async_le16() {
#if defined(__has_builtin) && __has_builtin(__builtin_amdgcn_s_wait_asynccnt)
  __builtin_amdgcn_s_wait_asynccnt(16);
#else
  asm volatile("s_wait_asynccnt 0x10" ::: "memory");
#endif
}
__device__ __forceinline__ void wait_async_le8() {
#if defined(__has_builtin) && __has_builtin(__builtin_amdgcn_s_wait_asynccnt)
  __builtin_amdgcn_s_wait_asynccnt(8);
#else
  asm volatile("s_wait_asynccnt 0x8" ::: "memory");
#endif
}
__device__ __forceinline__ void wait_async_le0() {
#if defined(__has_builtin) && __has_builtin(__builtin_amdgcn_s_wait_asynccnt)
  __builtin_amdgcn_s_wait_asynccnt(0);
#else
  asm volatile("s_wait_asynccnt 0x0" ::: "memory");
#endif
}

__global__ void __launch_bounds__(BLOCK)
snn_leaky2_kernel(const float* __restrict__ x,
                  const float* __restrict__ W1, const float* __restrict__ B1,
                  const float* __restrict__ W2, const float* __restrict__ B2,
                  float* __restrict__ out_spk, float* __restrict__ out_mem)
{
  __shared__ float smem[NBUF][TT * ROWF];   // 4 x 16KB input staging

  const int tid = threadIdx.x;
  const int b0  = blockIdx.x * BLOCK;

  // 1x1 layers fold to scalars; spk1 in {0,1} => cur2 is a select.
  const float w1  = W1[0];
  const float bb1 = B1[0];
  const float bb2 = B2[0];
  const float wb2 = W2[0] + bb2;

  // Per-thread global byte offset within a chunk (row = tid/32, lane4 = tid%32).
  const unsigned goff = ((unsigned)(tid >> 5)) * (BATCH * 4u)
                      + ((unsigned)(tid & 31)) * 16u;
  unsigned lbase[NBUF];
#pragma unroll
  for (int k = 0; k < NBUF; ++k)
    lbase[k] = lds_addr_of(&smem[k][0]) + (unsigned)tid * 16u;

  const float* xg = x + b0;

  // Prologue: 3 chunks in flight (24 async ops/wave, 48KB/block outstanding).
  async_chunk(xg,                            lbase[0], goff);
  async_chunk(xg + (size_t)TT * BATCH,       lbase[1], goff);
  async_chunk(xg + (size_t)2 * TT * BATCH,   lbase[2], goff);

  float m1 = 0.0f, m2 = 0.0f;
  const unsigned obase = (unsigned)(b0 + tid);

  for (int c = 0; c < NCHUNK; ++c) {
    // Retire chunk c: allow the (up to) two younger chunks to stay in flight.
    const int rem = NCHUNK - 1 - c;
    if (rem >= 2)      wait_async_le16();
    else if (rem == 1) wait_async_le8();
    else               wait_async_le0();
    __syncthreads();                      // chunk written by all waves -> visible

    const float* rb = smem[c & (NBUF - 1)];
    unsigned idx = obase + (unsigned)(c * TT) * (unsigned)BATCH;
#pragma unroll
    for (int tt = 0; tt < TT; ++tt) {
      float xv   = rb[tt * ROWF + tid];               // ds_load_b32, conflict-free
      float cur1 = __builtin_fmaf(xv, w1, bb1);
      float r1   = (m1 > 1.0f) ? 1.0f : 0.0f;         // reset from PREVIOUS mem1
      m1 = __builtin_fmaf(0.9f, m1, cur1) - r1;
      float cur2 = (m1 > 1.0f) ? wb2 : bb2;           // spk1*W2 + b2
      float r2   = (m2 > 1.0f) ? 1.0f : 0.0f;         // reset from PREVIOUS mem2
      m2 = __builtin_fmaf(0.9f, m2, cur2) - r2;
      float s2   = (m2 > 1.0f) ? 1.0f : 0.0f;

      __builtin_nontemporal_store(s2, out_spk + idx); // streamed once -> NT
      __builtin_nontemporal_store(m2, out_mem + idx);
      idx += (unsigned)BATCH;
    }
    __syncthreads();                      // reads done before buffer reuse

    if (c + 3 < NCHUNK)
      async_chunk(xg + (size_t)(c + 3) * TT * BATCH,
                  lbase[(c + 3) & (NBUF - 1)], goff);
  }
}

extern "C" void kernel_launch(void* const* d_in, const int* in_sizes, int n_in,
                              void* d_out, int out_size, void* d_ws, size_t ws_size,
                              hipStream_t stream) {
  const float* x  = (const float*)d_in[0];
  const float* W1 = (const float*)d_in[1];
  const float* b1 = (const float*)d_in[2];
  const float* W2 = (const float*)d_in[3];
  const float* b2 = (const float*)d_in[4];

  float* out_spk = (float*)d_out;                       // spk2_rec (T,B,1)
  float* out_mem = out_spk + (size_t)T_STEPS * BATCH;   // mem2_rec (T,B,1)

  dim3 grid(BATCH / BLOCK);   // 256 blocks
  dim3 block(BLOCK);          // 128 threads = 4 wave32
  snn_leaky2_kernel<<<grid, block, 0, stream>>>(x, W1, b1, W2, b2, out_spk, out_mem);
}